// HurstEstimator_37168646979587
// MI455X (gfx1250) — compile-verified
//
#include <hip/hip_runtime.h>
#include <hip/hip_bf16.h>
#include <math.h>

#define T_LEN    8192
#define NTHREADS 256
#define EPSF     1e-8f

typedef __attribute__((ext_vector_type(2))) float v2f;
typedef __attribute__((ext_vector_type(8))) float v8f;

#if defined(__AMDGCN__) && __has_builtin(__builtin_amdgcn_global_load_async_to_lds_b128)
#define HAVE_ASYNC 1
typedef int v4i_ __attribute__((vector_size(4 * sizeof(int))));
typedef __attribute__((address_space(1))) v4i_ as1_v4i;   // global b128 payload
typedef __attribute__((address_space(3))) v4i_ as3_v4i;   // LDS b128 payload
#else
#define HAVE_ASYNC 0
#endif

// order-preserving float<->uint keys so max/min can use integer LDS atomics
// (order-independent => deterministic across replays)
__device__ __forceinline__ unsigned f2ord(float f) {
  unsigned b = __float_as_uint(f);
  return (b & 0x80000000u) ? ~b : (b | 0x80000000u);
}
__device__ __forceinline__ float ord2f(unsigned u) {
  unsigned b = (u & 0x80000000u) ? (u & 0x7FFFFFFFu) : ~u;
  return __uint_as_float(b);
}

__global__ __launch_bounds__(NTHREADS)
void HurstEstimator_37168646979587_kernel(const float* __restrict__ xg,
                                          float* __restrict__ out) {
  __shared__ float    Pbuf[T_LEN];       // x streamed in, overwritten by prefix sums
  __shared__ float    scanbuf[NTHREADS];
  __shared__ float    tqbuf[NTHREADS];
  __shared__ float    sq256[32];         // sum(x^2) per 256-element block
  __shared__ float    meanbuf[32];
  __shared__ float    sumbuf[32];
  __shared__ float    rsbuf[32];
  __shared__ unsigned chmaxU[32];
  __shared__ unsigned chminU[32];
  __shared__ float    lrbuf[8];

  const int tid = threadIdx.x;
  const int row = blockIdx.x;
  const float* rowp = xg + (size_t)row * T_LEN;

  float xv[32];
  float tq = 0.f;

  // ---------------- Phase A+B: stream row (async global->LDS on CDNA5),
  // per-thread 32-element load, sumsq, local inclusive scan ----------------
#if HAVE_ASYNC
  #pragma unroll
  for (int it = 0; it < 8; ++it) {
    __builtin_amdgcn_global_load_async_to_lds_b128(
        (as1_v4i*)(rowp + tid * 32 + it * 4),
        (as3_v4i*)(&Pbuf[tid * 32 + it * 4]),
        0, 0);
  }
#if __has_builtin(__builtin_amdgcn_s_wait_asynccnt)
  __builtin_amdgcn_s_wait_asynccnt(0);
#else
  asm volatile("s_wait_asynccnt 0" ::: "memory");
#endif
  __syncthreads();
  #pragma unroll
  for (int it = 0; it < 8; ++it) {
    float4 v = *(const float4*)&Pbuf[tid * 32 + it * 4];
    xv[it * 4 + 0] = v.x; xv[it * 4 + 1] = v.y;
    xv[it * 4 + 2] = v.z; xv[it * 4 + 3] = v.w;
    tq += v.x * v.x + v.y * v.y + v.z * v.z + v.w * v.w;
  }
#else
  #pragma unroll
  for (int it = 0; it < 8; ++it) {
    float4 v = ((const float4*)rowp)[tid * 8 + it];
    xv[it * 4 + 0] = v.x; xv[it * 4 + 1] = v.y;
    xv[it * 4 + 2] = v.z; xv[it * 4 + 3] = v.w;
    tq += v.x * v.x + v.y * v.y + v.z * v.z + v.w * v.w;
  }
#endif

  #pragma unroll
  for (int j = 1; j < 32; ++j) xv[j] += xv[j - 1];  // local inclusive scan
  const float t = xv[31];
  scanbuf[tid] = t;
  tqbuf[tid] = tq;
  __syncthreads();

  // ---------------- Phase C: deterministic Hillis-Steele block scan -------
  for (int off = 1; off < NTHREADS; off <<= 1) {
    float add = (tid >= off) ? scanbuf[tid - off] : 0.f;
    __syncthreads();
    scanbuf[tid] += add;
    __syncthreads();
  }
  const float excl = scanbuf[tid] - t;

  // ---------------- Phase D: write inclusive prefix sums P (in place) -----
  #pragma unroll
  for (int it = 0; it < 8; ++it) {
    float4 w;
    w.x = xv[it * 4 + 0] + excl; w.y = xv[it * 4 + 1] + excl;
    w.z = xv[it * 4 + 2] + excl; w.w = xv[it * 4 + 3] + excl;
    *(float4*)&Pbuf[tid * 32 + it * 4] = w;
  }
  if (tid < 32) {  // fixed-order (deterministic) 256-block sumsq table
    float acc0 = 0.f;
    #pragma unroll
    for (int j = 0; j < 8; ++j) acc0 += tqbuf[tid * 8 + j];
    sq256[tid] = acc0;
  }
  __syncthreads();

  const int lane = tid & 31;
  const int wave = tid >> 5;
  const int half = lane >> 4;   // 0: lanes 0-15 (rows 0-7), 1: lanes 16-31 (rows 8-15)
  const int l16  = lane & 15;   // column within tile

  // ---------------- Phase E: per-scale R/S via WMMA rank-1 tiles ----------
  for (int k = 0; k < 5; ++k) {
    const int s  = 256 << k;
    const int nc = T_LEN / s;

    if (tid < 32) { chmaxU[tid] = 0u; chminU[tid] = 0xFFFFFFFFu; }
    if (tid < nc) {
      float hi = Pbuf[(tid + 1) * s - 1];
      float lo = (tid > 0) ? Pbuf[tid * s - 1] : 0.f;
      float sm = hi - lo;
      sumbuf[tid]  = sm;
      meanbuf[tid] = sm / (float)s;
    }
    __syncthreads();

    const int ntc  = (nc + 15) >> 4;   // column tiles (chunks)
    const int ntq  = s >> 4;           // row tiles (positions in chunk)
    const int ntot = ntc * ntq;

    // D = C + A*B : A[:,0] = -(pos+1) ramp, B[0,:] = chunk means, C = P tile.
    // Only A col 0 (documented layout) and B VGPR0/lanes0-15 (K=0 row) are
    // non-zero, so the rank-1 product is robust to B layout details.
    for (int tile = wave; tile < ntot; tile += 8) {
      const int ct   = tile % ntc;
      const int q    = tile / ntc;
      const int qoff = q << 4;
      const int ch   = ct * 16 + l16;
      const int chc  = (ch < nc) ? ch : (nc - 1);

      v2f a, b;
      a.x = (half == 0) ? -(float)(qoff + l16 + 1) : 0.f;
      a.y = 0.f;
      b.x = (half == 0) ? ((ch < nc) ? meanbuf[ch] : 0.f) : 0.f;
      b.y = 0.f;

      const float4* p4 = (const float4*)&Pbuf[chc * s + qoff + half * 8];
      float4 c0 = p4[0], c1 = p4[1];
      v8f cc;
      cc[0] = c0.x; cc[1] = c0.y; cc[2] = c0.z; cc[3] = c0.w;
      cc[4] = c1.x; cc[5] = c1.y; cc[6] = c1.z; cc[7] = c1.w;

      v8f d = __builtin_amdgcn_wmma_f32_16x16x4_f32(
          false, a, false, b, (short)0, cc, false, false);

      float mx = d[0], mn = d[0];
      #pragma unroll
      for (int r = 1; r < 8; ++r) { mx = fmaxf(mx, d[r]); mn = fminf(mn, d[r]); }
      if (ch < nc) {
        atomicMax(&chmaxU[ch], f2ord(mx));
        atomicMin(&chminU[ch], f2ord(mn));
      }
    }
    __syncthreads();

    if (tid < nc) {
      float R  = ord2f(chmaxU[tid]) - ord2f(chminU[tid]);
      float sm = sumbuf[tid];
      const int nb = s >> 8;
      float ssq = 0.f;
      for (int j = 0; j < nb; ++j) ssq += sq256[tid * nb + j];
      float var = (ssq - sm * sm / (float)s) / (float)(s - 1);
      var = fmaxf(var, 0.f);
      float Sd = fmaxf(sqrtf(var), EPSF);
      rsbuf[tid] = fmaxf(R / Sd, EPSF);
    }
    __syncthreads();
    if (tid == 0) {
      float acc = 0.f;
      for (int c2 = 0; c2 < nc; ++c2) acc += rsbuf[c2];
      acc /= (float)nc;
      lrbuf[k] = logf(fmaxf(acc, EPSF));
    }
    __syncthreads();
  }

  // ---------------- Phase F: log-log OLS slope, clip ----------------------
  if (tid == 0) {
    float ln[5], lm = 0.f;
    #pragma unroll
    for (int k = 0; k < 5; ++k) { ln[k] = logf((float)(256 << k)); lm += ln[k]; }
    lm *= 0.2f;
    float lrm = 0.f;
    #pragma unroll
    for (int k = 0; k < 5; ++k) lrm += lrbuf[k];
    lrm *= 0.2f;
    float num = 0.f, den = 0.f;
    #pragma unroll
    for (int k = 0; k < 5; ++k) {
      float lc = ln[k] - lm;
      num += lc * (lrbuf[k] - lrm);
      den += lc * lc;
    }
    den = fmaxf(den, 1e-8f);
    float H = num / den;
    H = fminf(fmaxf(H, 0.05f), 0.95f);
    out[row] = H;
  }
}

extern "C" void kernel_launch(void* const* d_in, const int* in_sizes, int n_in,
                              void* d_out, int out_size, void* d_ws, size_t ws_size,
                              hipStream_t stream) {
  const float* x = (const float*)d_in[0];
  float* out = (float*)d_out;
  const int B = in_sizes[0] / T_LEN;   // T = 8192 static per reference
  HurstEstimator_37168646979587_kernel<<<B, NTHREADS, 0, stream>>>(x, out);
}